// LSTMDiscriminator_73426760892456
// MI455X (gfx1250) — compile-verified
//
#include <hip/hip_runtime.h>
#include <math.h>

// ---------------------------------------------------------------------------
// LSTMDiscriminator for MI455X (gfx1250).
//
// The reference output is a single scalar computed from h[0] (track 0) only,
// and the LSTM recurrence never mixes tracks -> evolve just a 16-track tile
// (one WMMA M-dimension, tracks 0..15) through the 21 sequential steps with
// V_WMMA_F32_16X16X4_F32 (bit-exact fp32). One workgroup, latency-bound.
//
// Round-2 improvements:
//  * Gate-major N-tile assignment: wave w owns N-tiles {w, 8+w, 16+w, 24+w},
//    so acc0..acc3 = (i,f,g,o) gates for the same j-range. The LSTM cell
//    nonlinearity runs entirely in registers; c lives in 8 VGPRs/lane,
//    biases fold into 4 per-lane scalars hoisted out of the time loop.
//    The LDS gates buffer and the separate nonlinearity pass are gone.
//  * Depth-2 software pipeline on the B-tile global loads (fully unrolled
//    K loop) so WMMAs no longer sit behind s_wait_loadcnt 0 every issue.
// ---------------------------------------------------------------------------

typedef float v2f __attribute__((ext_vector_type(2)));
typedef float v8f __attribute__((ext_vector_type(8)));

#define EMB      64
#define HID      128
#define NKT      48           // K tiles: 192 / 4
#define NT       16           // tracks computed (WMMA tile height)
#define NTRACKS  32768
#define TOBS     9
#define TTOT     21

#define XH_S     196          // padded row stride for [x|h] (conflict-free A loads)

__device__ __forceinline__ float sigmoid_f(float x) {
    return 1.0f / (1.0f + expf(-x));
}

#define WMMA_F32X4(A, B, C) \
    __builtin_amdgcn_wmma_f32_16x16x4_f32(false, (A), false, (B), (short)0, (C), false, false)

extern "C" __global__ void __launch_bounds__(256)
lstm_disc_kernel(const float* __restrict__ observed,
                 const float* __restrict__ prediction,
                 const float* __restrict__ W_emb, const float* __restrict__ b_emb,
                 const float* __restrict__ W_ih,  const float* __restrict__ W_hh,
                 const float* __restrict__ b_ih,  const float* __restrict__ b_hh,
                 const float* __restrict__ W1, const float* __restrict__ b1,
                 const float* __restrict__ W2, const float* __restrict__ b2,
                 const float* __restrict__ W3, const float* __restrict__ b3,
                 float* __restrict__ out)
{
    __shared__ float sXH[NT * XH_S];   // per-track [x(64) | h(128)]
    __shared__ float sVX[NT], sVY[NT];
    __shared__ int   sMSK[NT];
    __shared__ float sB1[64], sB2[32];

    const int tid  = threadIdx.x;
    const int lane = tid & 31;
    const int wave = tid >> 5;         // 0..7

    // ---- per-lane fixed roles -------------------------------------------
    const int mA = lane & 15;          // A operand: row M
    const int ka = (lane >> 4) << 1;   // A operand: K base 0 or 2
    const int kb = lane >> 4;          // B operand: K row 0 or 1 (pair +2)
    const int hl = lane >> 4;          // which 8-track half this lane's D rows hit
    const int jn = (wave << 4) + (lane & 15);   // cell index j in [0,128)

    // Weight row bases for the four gate classes (torch order i,f,g,o)
    const float* Wih_i = W_ih + (size_t)(jn)           * EMB;
    const float* Wih_f = W_ih + (size_t)(HID   + jn)   * EMB;
    const float* Wih_g = W_ih + (size_t)(2*HID + jn)   * EMB;
    const float* Wih_o = W_ih + (size_t)(3*HID + jn)   * EMB;
    const float* Whh_i = W_hh + (size_t)(jn)           * HID;
    const float* Whh_f = W_hh + (size_t)(HID   + jn)   * HID;
    const float* Whh_g = W_hh + (size_t)(2*HID + jn)   * HID;
    const float* Whh_o = W_hh + (size_t)(3*HID + jn)   * HID;

    // Biases (constant over time and tracks for this lane's j)
    const float bias_i = b_ih[jn]          + b_hh[jn];
    const float bias_f = b_ih[HID + jn]    + b_hh[HID + jn];
    const float bias_g = b_ih[2*HID + jn]  + b_hh[2*HID + jn];
    const float bias_o = b_ih[3*HID + jn]  + b_hh[3*HID + jn];

    // Cell state lives in registers: cell (track = hl*8 + r, j = jn)
    float creg[8];
    #pragma unroll
    for (int r = 0; r < 8; ++r) creg[r] = 0.0f;

    // h = 0 in LDS
    for (int u = tid; u < NT * HID; u += 256) {
        int m = u >> 7, j = u & 127;
        sXH[m * XH_S + EMB + j] = 0.0f;
    }

    // B-tile fetch for K-slice kt into 4 gate-class v2f registers
    auto loadB = [&](int kt, v2f& bi, v2f& bf, v2f& bg, v2f& bo) {
        const int k0 = kt * 4;
        if (k0 < EMB) {
            const int k = k0 + kb;
            bi.x = Wih_i[k]; bi.y = Wih_i[k + 2];
            bf.x = Wih_f[k]; bf.y = Wih_f[k + 2];
            bg.x = Wih_g[k]; bg.y = Wih_g[k + 2];
            bo.x = Wih_o[k]; bo.y = Wih_o[k + 2];
        } else {
            const int k = k0 - EMB + kb;
            bi.x = Whh_i[k]; bi.y = Whh_i[k + 2];
            bf.x = Whh_f[k]; bf.y = Whh_f[k + 2];
            bg.x = Whh_g[k]; bg.y = Whh_g[k + 2];
            bo.x = Whh_o[k]; bo.y = Whh_o[k + 2];
        }
    };

    // s == -1 : start_enc tag step; s in [0, 19] : velocity steps
    for (int s = -1; s < TTOT - 1; ++s) {
        // ---- velocity + visibility mask ----
        if (s >= 0 && tid < NT) {
            int m = tid;
            const float* p1 = (s < TOBS)
                ? observed   + ((size_t)s          * NTRACKS + m) * 2
                : prediction + ((size_t)(s - TOBS) * NTRACKS + m) * 2;
            int s2 = s + 1;
            const float* p2 = (s2 < TOBS)
                ? observed   + ((size_t)s2          * NTRACKS + m) * 2
                : prediction + ((size_t)(s2 - TOBS) * NTRACKS + m) * 2;
            float o1x = p1[0], o1y = p1[1];
            float o2x = p2[0], o2y = p2[1];
            int vis = !(isnan(o1x) || isnan(o2x));   // reference tests comp 0 only
            sVX[m]  = vis ? (o2x - o1x) : 0.0f;
            sVY[m]  = vis ? (o2y - o1y) : 0.0f;
            sMSK[m] = vis;
        }
        if (s < 0 && tid < NT) sMSK[tid] = 1;
        __syncthreads();

        // ---- build x part of [x|h] ----
        if (s < 0) {
            for (int u = tid; u < NT * EMB; u += 256) {
                int m = u >> 6, e = u & 63;
                sXH[m * XH_S + e] = (e == EMB - 2) ? 1.0f : 0.0f;
            }
        } else {
            for (int u = tid; u < NT * EMB; u += 256) {
                int m = u >> 6, e = u & 63;
                float v = 0.0f;
                if (e < EMB - 2) {
                    v = fmaxf(0.0f, 4.0f * (sVX[m] * W_emb[2*e] + sVY[m] * W_emb[2*e+1])
                                    + b_emb[e]);
                }
                sXH[m * XH_S + e] = v;   // slots 62,63 = appended zeros
            }
        }
        __syncthreads();

        // ---- gates via V_WMMA_F32_16X16X4_F32, depth-2 pipelined B loads ----
        v8f acc_i = {}, acc_f = {}, acc_g = {}, acc_o = {};
        {
            const float* arow = &sXH[mA * XH_S];
            v2f Bi0, Bf0, Bg0, Bo0, Bi1, Bf1, Bg1, Bo1;
            loadB(0, Bi0, Bf0, Bg0, Bo0);
            loadB(1, Bi1, Bf1, Bg1, Bo1);
            #pragma unroll
            for (int kt = 0; kt < NKT; ++kt) {
                v2f Ni, Nf, Ng, No;
                if (kt + 2 < NKT) loadB(kt + 2, Ni, Nf, Ng, No);
                else { Ni = Bi1; Nf = Bf1; Ng = Bg1; No = Bo1; }
                v2f a;
                const int k0 = kt * 4;
                a.x = arow[k0 + ka];
                a.y = arow[k0 + ka + 1];
                acc_i = WMMA_F32X4(a, Bi0, acc_i);
                acc_f = WMMA_F32X4(a, Bf0, acc_f);
                acc_g = WMMA_F32X4(a, Bg0, acc_g);
                acc_o = WMMA_F32X4(a, Bo0, acc_o);
                Bi0 = Bi1; Bf0 = Bf1; Bg0 = Bg1; Bo0 = Bo1;
                Bi1 = Ni;  Bf1 = Nf;  Bg1 = Ng;  Bo1 = No;
            }
        }

        // ---- LSTM cell update, fully in registers ----
        // D layout: VGPR r -> track hl*8 + r; column -> j = jn
        float hnew[8];
        int   upd[8];
        #pragma unroll
        for (int r = 0; r < 8; ++r) {
            const int trk = hl * 8 + r;
            upd[r] = sMSK[trk];
            if (upd[r]) {
                float iv = sigmoid_f(acc_i[r] + bias_i);
                float fv = sigmoid_f(acc_f[r] + bias_f);
                float gv = tanhf    (acc_g[r] + bias_g);
                float ov = sigmoid_f(acc_o[r] + bias_o);
                float cn = fv * creg[r] + iv * gv;
                creg[r]  = cn;
                hnew[r]  = ov * tanhf(cn);
            }
        }
        __syncthreads();   // all WMMA reads of sXH complete

        #pragma unroll
        for (int r = 0; r < 8; ++r) {
            if (upd[r]) sXH[(hl * 8 + r) * XH_S + EMB + jn] = hnew[r];
        }
        // visibility of h stores is guaranteed by next iteration's barriers
    }
    __syncthreads();       // final h visible for the classifier

    // ---- real_classifier on h of track 0: 128 -> 64 -> 32 -> 1, ReLU each ----
    if (tid < 64) {
        float a = b1[tid];
        const float* h0 = &sXH[EMB];
        const float* w  = &W1[tid * HID];
        #pragma unroll 8
        for (int j = 0; j < HID; ++j) a += h0[j] * w[j];
        sB1[tid] = fmaxf(a, 0.0f);
    }
    __syncthreads();
    if (tid < 32) {
        float a = b2[tid];
        const float* w = &W2[tid * 64];
        #pragma unroll 8
        for (int j = 0; j < 64; ++j) a += sB1[j] * w[j];
        sB2[tid] = fmaxf(a, 0.0f);
    }
    __syncthreads();
    if (tid == 0) {
        float a = b3[0];
        #pragma unroll
        for (int j = 0; j < 32; ++j) a += sB2[j] * W3[j];
        out[0] = fmaxf(a, 0.0f);
    }
}

extern "C" void kernel_launch(void* const* d_in, const int* in_sizes, int n_in,
                              void* d_out, int out_size, void* d_ws, size_t ws_size,
                              hipStream_t stream) {
    (void)in_sizes; (void)n_in; (void)out_size; (void)d_ws; (void)ws_size;
    lstm_disc_kernel<<<1, 256, 0, stream>>>(
        (const float*)d_in[0],  (const float*)d_in[1],
        (const float*)d_in[2],  (const float*)d_in[3],
        (const float*)d_in[4],  (const float*)d_in[5],
        (const float*)d_in[6],  (const float*)d_in[7],
        (const float*)d_in[8],  (const float*)d_in[9],
        (const float*)d_in[10], (const float*)d_in[11],
        (const float*)d_in[12], (const float*)d_in[13],
        (float*)d_out);
}